// CLAPP_layer_16295105921261
// MI455X (gfx1250) — compile-verified
//
#include <hip/hip_runtime.h>
#include <hip/hip_bf16.h>

typedef float v2f __attribute__((ext_vector_type(2)));
typedef float v8f __attribute__((ext_vector_type(8)));

#define HN 8192
#define NCHUNK 8
#define KCHUNK (HN / NCHUNK)   // 1024

__device__ __forceinline__ float surrogate(float x) {
    const float pi = 3.14159265358979323846f;
    float px = pi * x;
    return 1.0f / (pi * (1.0f + px * px));
}

// ---------------------------------------------------------------------------
// Split-K matvec via V_WMMA_F32_16X16X4_F32.
// Wave w handles 16 rows (rg) x K-chunk (ck); writes partial row sums to
// part[ck*N + row]. A-tile layout per ISA 7.12.2 (32-bit A 16x4):
//   lane L: m = L&15, VGPR0 = K = 2*(L>>4), VGPR1 = K+1  -> one b64 load/lane.
// B broadcast across N (vector matvec), so only the per-lane K mapping matters.
// D column N=0: lane 0 holds rows 0..7 in c[0..7], lane 16 holds rows 8..15.
// ---------------------------------------------------------------------------
__global__ void clapp_matvec_wmma(const float* __restrict__ W,
                                  const float* __restrict__ x,
                                  float* __restrict__ part,
                                  int N) {
    const int lane = threadIdx.x & 31;
    const int wave = blockIdx.x * (blockDim.x >> 5) + (threadIdx.x >> 5);
    const int nRG  = N >> 4;
    const int rb   = (wave % nRG) << 4;
    const int ck   = wave / nRG;
    const int k0   = ck * KCHUNK;
    const int m    = lane & 15;
    const int kk   = (lane >> 4) << 1;           // 0 or 2
    const float* Wrow = W + (size_t)(rb + m) * N;

    v8f c = {};
    for (int k = k0; k < k0 + KCHUNK; k += 4) {
        const int j = k + kk;
        v2f a = *(const v2f*)(Wrow + j);
        v2f b; b.x = x[j]; b.y = x[j + 1];
        c = __builtin_amdgcn_wmma_f32_16x16x4_f32(
                false, a, false, b, (short)0, c, false, false);
    }
    if (m == 0) {
        const int half = (lane >> 4) << 3;       // 0 or 8
        float* dst = part + (size_t)ck * N + rb + half;
#pragma unroll
        for (int t = 0; t < 8; ++t) dst[t] = c[t];
    }
}

// ---------------------------------------------------------------------------
// Fused pass over a HxH matrix:
//   partials[ck][i] = sum_{j in chunk} W[i,j]*bvec[j]    (WMMA accumulate)
//   Wout[i,j]       = W[i,j] + (LR*bf)*rowv[i]*colv[j]   (rank-1 update)
// Single read + single write of the 256MB matrix.
// ---------------------------------------------------------------------------
__global__ void clapp_fused_update_wmma(const float* __restrict__ W,
                                        const float* __restrict__ bvec,
                                        const float* __restrict__ rowv,
                                        const float* __restrict__ colv,
                                        const float* __restrict__ bfp,
                                        float* __restrict__ Wout,
                                        float* __restrict__ part,
                                        int N) {
    const int lane = threadIdx.x & 31;
    const int wave = blockIdx.x * (blockDim.x >> 5) + (threadIdx.x >> 5);
    const int nRG  = N >> 4;
    const int rb   = (wave % nRG) << 4;
    const int ck   = wave / nRG;
    const int k0   = ck * KCHUNK;
    const int m    = lane & 15;
    const int kk   = (lane >> 4) << 1;
    const float rv = (2e-6f * bfp[0]) * rowv[rb + m];   // LR * bf * rowv[i]
    const float* Wrow = W    + (size_t)(rb + m) * N;
    float*       Orow = Wout + (size_t)(rb + m) * N;

    v8f c = {};
    for (int k = k0; k < k0 + KCHUNK; k += 4) {
        const int j = k + kk;
        v2f a = *(const v2f*)(Wrow + j);
        v2f b; b.x = bvec[j]; b.y = bvec[j + 1];
        c = __builtin_amdgcn_wmma_f32_16x16x4_f32(
                false, a, false, b, (short)0, c, false, false);
        v2f o;
        o.x = a.x + rv * colv[j];
        o.y = a.y + rv * colv[j + 1];
        *(v2f*)(Orow + j) = o;
    }
    if (m == 0) {
        const int half = (lane >> 4) << 3;
        float* dst = part + (size_t)ck * N + rb + half;
#pragma unroll
        for (int t = 0; t < 8; ++t) dst[t] = c[t];
    }
}

// Reduce fc partials (fixed order -> deterministic), then LIF step.
__global__ void clapp_lif(const float* __restrict__ P0,
                          const float* __restrict__ fc_b,
                          const float* __restrict__ mem,
                          float* __restrict__ cur,
                          float* __restrict__ spk,
                          float* __restrict__ mem_out,
                          int H) {
    int i = blockIdx.x * blockDim.x + threadIdx.x;
    if (i >= H) return;
    float c = fc_b[i];
#pragma unroll
    for (int k = 0; k < NCHUNK; ++k) c += P0[(size_t)k * H + i];
    cur[i] = c;
    float mn = 0.95f * mem[i] + c;
    float sp = (mn > 1.0f) ? 1.0f : 0.0f;
    spk[i]     = sp;
    mem_out[i] = mn - sp;
}

// Deterministic sum(spk) -> nspk (single block, LDS tree).
__global__ void clapp_spk_sum(const float* __restrict__ spk,
                              float* __restrict__ nspk, int H) {
    __shared__ float red[256];
    float v = 0.0f;
    for (int i = threadIdx.x; i < H; i += 256) v += spk[i];
    red[threadIdx.x] = v;
    __syncthreads();
    for (int s = 128; s > 0; s >>= 1) {
        if ((int)threadIdx.x < s) red[threadIdx.x] += red[threadIdx.x + s];
        __syncthreads();
    }
    if (threadIdx.x == 0) nspk[0] = red[0];
}

// Reduce retro/pred partials; build r, s and feedback_new.
__global__ void clapp_vectors(const float* __restrict__ P1,   // retro partials
                              const float* __restrict__ P2,   // pred partials
                              const float* __restrict__ cur,
                              const float* __restrict__ cur_prev,
                              const float* __restrict__ feedback,
                              const float* __restrict__ bfp,
                              const float* __restrict__ nspk,
                              float* __restrict__ r,
                              float* __restrict__ s,
                              float* __restrict__ fb_new,
                              int H) {
    int i = blockIdx.x * blockDim.x + threadIdx.x;
    if (i >= H) return;
    float retro = 0.0f, pred = 0.0f;
#pragma unroll
    for (int k = 0; k < NCHUNK; ++k) {
        retro += P1[(size_t)k * H + i];
        pred  += P2[(size_t)k * H + i];
    }
    float bf   = bfp[0];
    float lrbf = 2e-6f * bf;
    r[i]      = lrbf * feedback[i] * surrogate(cur[i]);
    s[i]      = bf * retro * surrogate(cur_prev[i]);
    fb_new[i] = pred + lrbf * feedback[i] * nspk[0];
}

// fc_w_new = fc_w + r[i]*inp[j] + s[i]*prev_inp[j]  (pure stream, float4).
__global__ void clapp_fc_update(const float* __restrict__ fc_w,
                                const float* __restrict__ r,
                                const float* __restrict__ s,
                                const float* __restrict__ inp,
                                const float* __restrict__ pinp,
                                float* __restrict__ out,
                                int I) {
    const int row = blockIdx.y;
    const int col = (blockIdx.x * blockDim.x + threadIdx.x) << 2;
    const float rv = r[row], sv = s[row];
    const size_t off = (size_t)row * I + col;
    const float4 w = *(const float4*)(fc_w + off);
    const float4 a = *(const float4*)(inp + col);
    const float4 b = *(const float4*)(pinp + col);
    float4 o;
    o.x = w.x + rv * a.x + sv * b.x;
    o.y = w.y + rv * a.y + sv * b.y;
    o.z = w.z + rv * a.z + sv * b.z;
    o.w = w.w + rv * a.w + sv * b.w;
    *(float4*)(out + off) = o;
}

extern "C" void kernel_launch(void* const* d_in, const int* in_sizes, int n_in,
                              void* d_out, int out_size, void* d_ws, size_t ws_size,
                              hipStream_t stream) {
    const int H = HN, I = HN;
    const float* inp      = (const float*)d_in[0];
    const float* bf       = (const float*)d_in[1];
    const float* fc_w     = (const float*)d_in[2];
    const float* fc_b     = (const float*)d_in[3];
    const float* pred_w   = (const float*)d_in[4];
    const float* retro_w  = (const float*)d_in[5];
    const float* feedback = (const float*)d_in[6];
    const float* mem      = (const float*)d_in[7];
    const float* cur_prev = (const float*)d_in[8];
    const float* prev_inp = (const float*)d_in[9];

    float* out         = (float*)d_out;
    float* spk         = out;                              // [H]
    float* mem_out     = out + H;                          // [H]
    float* fc_w_new    = out + 2 * (size_t)H;              // [H,I]
    float* pred_w_new  = fc_w_new + (size_t)H * I;         // [H,H]
    float* retro_w_new = pred_w_new + (size_t)H * H;       // [H,H]
    float* fb_new      = retro_w_new + (size_t)H * H;      // [H]

    float* ws   = (float*)d_ws;
    float* P0   = ws;                                      // [NCHUNK,H] fc partials
    float* P1   = P0 + (size_t)NCHUNK * H;                 // [NCHUNK,H] retro partials
    float* P2   = P1 + (size_t)NCHUNK * H;                 // [NCHUNK,H] pred partials
    float* cur  = P2 + (size_t)NCHUNK * H;                 // [H]
    float* rv   = cur + H;                                 // [H]
    float* sv   = rv + H;                                  // [H]
    float* nspk = sv + H;                                  // [1]

    const int waves  = (H >> 4) * NCHUNK;                  // 4096
    const int mvBlks = waves / 8;                          // 256 thr = 8 waves

    // 1) cur partials = fc_w @ inp (split-K WMMA)
    clapp_matvec_wmma<<<mvBlks, 256, 0, stream>>>(fc_w, inp, P0, I);
    // 2) reduce + LIF -> cur, spk, mem_out
    clapp_lif<<<H / 256, 256, 0, stream>>>(P0, fc_b, mem, cur, spk, mem_out, H);
    // 3) nspk = sum(spk), deterministic
    clapp_spk_sum<<<1, 256, 0, stream>>>(spk, nspk, H);
    // 4) retro pass: acc=retro_w@spk, retro_w_new = retro_w + lrbf*outer(spk,feedback)
    clapp_fused_update_wmma<<<mvBlks, 256, 0, stream>>>(
        retro_w, spk, /*rowv=*/spk, /*colv=*/feedback, bf, retro_w_new, P1, H);
    // 5) pred pass: acc=pred_w@spk, pred_w_new = pred_w + lrbf*outer(feedback,spk)
    clapp_fused_update_wmma<<<mvBlks, 256, 0, stream>>>(
        pred_w, spk, /*rowv=*/feedback, /*colv=*/spk, bf, pred_w_new, P2, H);
    // 6) reduce retro/pred partials -> r, s, feedback_new
    clapp_vectors<<<H / 256, 256, 0, stream>>>(
        P1, P2, cur, cur_prev, feedback, bf, nspk, rv, sv, fb_new, H);
    // 7) fc_w_new = fc_w + outer(r, inp) + outer(s, prev_inp)
    dim3 g6(I / 1024, H);
    clapp_fc_update<<<g6, 256, 0, stream>>>(fc_w, rv, sv, inp, prev_inp, fc_w_new, I);
}